// RTMScore_14508399526693
// MI455X (gfx1250) — compile-verified
//
#include <hip/hip_runtime.h>
#include <math.h>

typedef __attribute__((ext_vector_type(2))) float v2f;
typedef __attribute__((ext_vector_type(8))) float v8f;

#define C_IN   128
#define HID    256
#define TWO_C  256
#define TILE_M 64
#define NG     10

// workspace float offsets
#define WS_SCALE 0                 // 256
#define WS_SHIFT 256               // 256
#define WS_WCATP 512               // 128*32 float2: packed [W_pi|W_sigma|W_mu|0], rows (2k,2k+1) interleaved
#define WS_BCAT  8704              // 32
#define WS_W1P   8736              // 128*256 float2: W1 packed in B-operand order
#define WS_PART  74272             // numWG * 512 (256 sum + 256 sumsq per WG)

// XOR swizzle so WMMA A-operand LDS reads (16 rows, same k) hit 16 distinct banks.
__device__ __forceinline__ int sw_idx(int row, int k) {
  return row * 256 + (k ^ ((row & 15) << 2));
}

__device__ __forceinline__ void gather_C(float* cbuf,
    const float* __restrict__ lig, const float* __restrict__ prot,
    const int* __restrict__ pli, const int* __restrict__ ppi,
    long long wgbase, long long P, int tid, int nthr) {
  // 64 rows x 64 groups of float4
  for (int idx = tid; idx < TILE_M * 64; idx += nthr) {
    int r = idx >> 6;
    int c = (idx & 63) * 4;
    long long p = wgbase + r;
    float4 v = make_float4(0.f, 0.f, 0.f, 0.f);
    if (p < P) {
      if (c < C_IN) {
        v = *(const float4*)(lig + (long long)pli[p] * C_IN + c);
      } else {
        v = *(const float4*)(prot + (long long)ppi[p] * C_IN + (c - C_IN));
      }
    }
    // xor constant has low 2 bits clear and c % 4 == 0 -> contiguous float4 slot
    *(float4*)&cbuf[sw_idx(r, c)] = v;
  }
}

// One wave computes a 16x256 tile of h = C @ W1 with fp32 WMMA (K=4 per step).
// W1p is pre-packed: W1p[kp*256 + n] = (W1[2kp][n], W1[2kp+1][n]) -> one b64 per B fragment.
__device__ __forceinline__ void gemm_tile(const float2* __restrict__ W1p,
                                          const float* cbuf,
                                          int wv, int lo, int hi, v8f acc[16]) {
  const v8f vzero = {0.f,0.f,0.f,0.f,0.f,0.f,0.f,0.f};
#pragma unroll
  for (int nt = 0; nt < 16; ++nt) acc[nt] = vzero;
  for (int k0 = 0; k0 < TWO_C; k0 += 4) {
    int ka = k0 + 2 * hi;                 // A reg0: K=k0+2*hi ; reg1: +1
    int kp = (k0 >> 1) + hi;              // packed-pair row index
    int ab = sw_idx(wv * 16 + lo, ka);    // ka even, swizzle const mult of 4
    v2f a; a.x = cbuf[ab]; a.y = cbuf[ab + 1];
    const float2* wrow = W1p + (long long)kp * HID;
#pragma unroll
    for (int nt = 0; nt < 16; ++nt) {
      float2 pr = wrow[nt * 16 + lo];
      v2f b; b.x = pr.x; b.y = pr.y;
      acc[nt] = __builtin_amdgcn_wmma_f32_16x16x4_f32(
          false, a, false, b, (short)0, acc[nt], false, false);
    }
  }
}

// Pack W1 [256][256] into B-operand order: one-shot, amortized over all tiles.
__global__ void k0_pack_w1(const float* __restrict__ W1, float* __restrict__ ws) {
  int idx = blockIdx.x * blockDim.x + threadIdx.x;   // 0..32767
  int kp = idx >> 8;
  int n  = idx & 255;
  float2 v;
  v.x = W1[(2 * kp)     * HID + n];
  v.y = W1[(2 * kp + 1) * HID + n];
  *(float2*)&ws[WS_W1P + (long long)idx * 2] = v;
}

__global__ __launch_bounds__(128) void k1_gemm_stats(
    const float* __restrict__ lig, const float* __restrict__ prot,
    const int* __restrict__ pli, const int* __restrict__ ppi,
    const float* __restrict__ b1,
    float* __restrict__ ws, long long P) {
  __shared__ __align__(16) float cbuf[TILE_M * 256];
  int tid = threadIdx.x;
  int wv = tid >> 5, lane = tid & 31, lo = lane & 15, hi = lane >> 4;
  long long wgbase = (long long)blockIdx.x * TILE_M;

  gather_C(cbuf, lig, prot, pli, ppi, wgbase, P, tid, blockDim.x);
  __syncthreads();

  v8f acc[16];
  gemm_tile((const float2*)(ws + WS_W1P), cbuf, wv, lo, hi, acc);

  __syncthreads();                 // done reading C, safe to reuse LDS
  float* partS = cbuf;             // [4][256]
  float* partQ = cbuf + 1024;      // [4][256]
#pragma unroll
  for (int nt = 0; nt < 16; ++nt) {
    int col = nt * 16 + lo;
    float bc = b1[col];
    float s = 0.f, q = 0.f;
#pragma unroll
    for (int r = 0; r < 8; ++r) {
      long long p = wgbase + wv * 16 + r + 8 * hi;   // D row of this element
      float m = (p < P) ? 1.0f : 0.0f;
      float v = (acc[nt][r] + bc) * m;
      s += v; q += v * v;
    }
    s += __shfl_xor(s, 16, 32);
    q += __shfl_xor(q, 16, 32);
    if (hi == 0) { partS[wv * 256 + col] = s; partQ[wv * 256 + col] = q; }
  }
  __syncthreads();
  for (int c = tid; c < 256; c += blockDim.x) {
    float S = partS[c] + partS[256 + c] + partS[512 + c] + partS[768 + c];
    float Q = partQ[c] + partQ[256 + c] + partQ[512 + c] + partQ[768 + c];
    ws[WS_PART + (long long)blockIdx.x * 512 + c] = S;
    ws[WS_PART + (long long)blockIdx.x * 512 + 256 + c] = Q;
  }
}

__device__ __forceinline__ float wcat_elem(const float* __restrict__ Wpi,
                                           const float* __restrict__ Wsig,
                                           const float* __restrict__ Wmu,
                                           int k, int n) {
  if (n < 10) return Wpi[k * NG + n];
  if (n < 20) return Wsig[k * NG + (n - 10)];
  if (n < 30) return Wmu[k * NG + (n - 20)];
  return 0.f;
}

__global__ void k2_finalize(const float* __restrict__ gamma,
                            const float* __restrict__ beta,
                            const float* __restrict__ Wpi, const float* __restrict__ bpi,
                            const float* __restrict__ Wsig, const float* __restrict__ bsig,
                            const float* __restrict__ Wmu, const float* __restrict__ bmu,
                            float* __restrict__ ws, int numWG, long long P) {
  int c = threadIdx.x;             // 256 threads, 1 block
  float S = 0.f, Q = 0.f;
  for (int w = 0; w < numWG; ++w) {
    S += ws[WS_PART + (long long)w * 512 + c];
    Q += ws[WS_PART + (long long)w * 512 + 256 + c];
  }
  float inv = 1.0f / (float)P;
  float mean = S * inv;
  float var = Q * inv - mean * mean;        // biased var, matches jnp
  float sc = gamma[c] * rsqrtf(var + 1e-5f);
  ws[WS_SCALE + c] = sc;
  ws[WS_SHIFT + c] = beta[c] - mean * sc;
  // pack Wcat [256][32] = [W_pi | W_sigma | W_mu | 0] in B-operand pair order
  if (c < 128) {
    for (int n = 0; n < 32; ++n) {
      float a0 = wcat_elem(Wpi, Wsig, Wmu, 2 * c,     n);
      float a1 = wcat_elem(Wpi, Wsig, Wmu, 2 * c + 1, n);
      ws[WS_WCATP + (long long)(c * 32 + n) * 2]     = a0;
      ws[WS_WCATP + (long long)(c * 32 + n) * 2 + 1] = a1;
    }
  }
  if (c < 32) {
    float b;
    if (c < 10)      b = bpi[c];
    else if (c < 20) b = bsig[c - 10];
    else if (c < 30) b = bmu[c - 20];
    else             b = 0.f;
    ws[WS_BCAT + c] = b;
  }
}

__global__ __launch_bounds__(128) void k3_gemm_heads(
    const float* __restrict__ lig, const float* __restrict__ prot,
    const int* __restrict__ pli, const int* __restrict__ ppi,
    const int* __restrict__ pb,
    const float* __restrict__ b1,
    const float* __restrict__ ws, float* __restrict__ out, long long P,
    long long o_sigma, long long o_mu, long long o_pb) {
  __shared__ __align__(16) float cbuf[TILE_M * 256];
  int tid = threadIdx.x;
  int wv = tid >> 5, lane = tid & 31, lo = lane & 15, hi = lane >> 4;
  long long wgbase = (long long)blockIdx.x * TILE_M;

  gather_C(cbuf, lig, prot, pli, ppi, wgbase, P, tid, blockDim.x);
  __syncthreads();

  v8f acc[16];
  gemm_tile((const float2*)(ws + WS_W1P), cbuf, wv, lo, hi, acc);

  // BatchNorm + ELU in registers; write e back into own rows of LDS in A layout.
  const float* scale = ws + WS_SCALE;
  const float* shift = ws + WS_SHIFT;
#pragma unroll
  for (int nt = 0; nt < 16; ++nt) {
    int col = nt * 16 + lo;
    float sc = scale[col], sh = shift[col], bc = b1[col];
#pragma unroll
    for (int r = 0; r < 8; ++r) {
      float v = (acc[nt][r] + bc) * sc + sh;
      float e = v > 0.f ? v : expm1f(v);
      int row = wv * 16 + r + 8 * hi;      // own wave's rows only
      cbuf[sw_idx(row, col)] = e;
    }
  }
  // heads: e(16x256) @ Wcat(256x32) -- two more WMMA accumulators per wave
  const float2* wcatp = (const float2*)(ws + WS_WCATP);
  const v8f vzero = {0.f,0.f,0.f,0.f,0.f,0.f,0.f,0.f};
  v8f p0 = vzero, p1 = vzero;
  for (int k0 = 0; k0 < TWO_C; k0 += 4) {
    int ka = k0 + 2 * hi;
    int kp = (k0 >> 1) + hi;
    int ab = sw_idx(wv * 16 + lo, ka);
    v2f a; a.x = cbuf[ab]; a.y = cbuf[ab + 1];
    float2 pr0 = wcatp[kp * 32 + lo];
    float2 pr1 = wcatp[kp * 32 + 16 + lo];
    v2f b0; b0.x = pr0.x; b0.y = pr0.y;
    v2f b1v; b1v.x = pr1.x; b1v.y = pr1.y;
    p0 = __builtin_amdgcn_wmma_f32_16x16x4_f32(false, a, false, b0, (short)0, p0, false, false);
    p1 = __builtin_amdgcn_wmma_f32_16x16x4_f32(false, a, false, b1v, (short)0, p1, false, false);
  }
  const float* bcat = ws + WS_BCAT;
  float bb0 = bcat[lo], bb1 = bcat[16 + lo];
  __syncthreads();                 // all waves done reading e before reuse
  float* outT = cbuf;              // [64][32]
#pragma unroll
  for (int r = 0; r < 8; ++r) {
    int row = wv * 16 + r + 8 * hi;
    outT[row * 32 + lo]      = p0[r] + bb0;
    outT[row * 32 + 16 + lo] = p1[r] + bb1;
  }
  __syncthreads();
  if (tid < TILE_M) {
    long long p = wgbase + tid;
    if (p < P) {
      float lg[NG];
      float mx = -3.0e38f;
      for (int j = 0; j < NG; ++j) { lg[j] = outT[tid * 32 + j]; mx = fmaxf(mx, lg[j]); }
      float ssum = 0.f;
      for (int j = 0; j < NG; ++j) { lg[j] = __expf(lg[j] - mx); ssum += lg[j]; }
      float rs = 1.0f / ssum;
      for (int j = 0; j < NG; ++j) out[p * NG + j] = lg[j] * rs;
      for (int j = 0; j < NG; ++j) {
        float x = outT[tid * 32 + 10 + j];
        out[o_sigma + p * NG + j] = (x > 0.f ? x : expm1f(x)) + 1.1f;
      }
      for (int j = 0; j < NG; ++j) {
        float x = outT[tid * 32 + 20 + j];
        out[o_mu + p * NG + j] = (x > 0.f ? x : expm1f(x)) + 1.0f;
      }
      out[o_pb + p] = (float)pb[p];
    }
  }
}

__global__ void k4_dist(const float* __restrict__ lpos, const float* __restrict__ ppos,
                        const int* __restrict__ pli, const int* __restrict__ ppi,
                        float* __restrict__ out, long long o_dist, long long P) {
  long long p = (long long)blockIdx.x * blockDim.x + threadIdx.x;
  if (p >= P) return;
  int li = pli[p], pi = ppi[p];
  float lx = lpos[li * 3], ly = lpos[li * 3 + 1], lz = lpos[li * 3 + 2];
  const float* pr = ppos + (long long)pi * 72;   // 24*3 contiguous floats
  float best = 3.4e38f;
#pragma unroll
  for (int a = 0; a < 24; ++a) {
    float dx = lx - pr[a * 3], dy = ly - pr[a * 3 + 1], dz = lz - pr[a * 3 + 2];
    best = fminf(best, dx * dx + dy * dy + dz * dz);
  }
  out[o_dist + p] = sqrtf(best);   // min of sqrt == sqrt of min
}

__global__ void k5_atom_types(const float* __restrict__ lig,
                              const float* __restrict__ Wat, const float* __restrict__ bat,
                              float* __restrict__ out, long long o_at, long long n) {
  long long idx = (long long)blockIdx.x * blockDim.x + threadIdx.x;
  if (idx >= n) return;
  long long i = idx / 17; int t = (int)(idx % 17);
  float acc = bat[t];
  for (int k = 0; k < C_IN; ++k) acc += lig[i * C_IN + k] * Wat[k * 17 + t];
  out[o_at + idx] = acc;
}

__global__ void k6_bond_types(const float* __restrict__ lig,
                              const long long* __restrict__ esrc,
                              const long long* __restrict__ edst,
                              const float* __restrict__ Wbt, const float* __restrict__ bbt,
                              float* __restrict__ out, long long o_bt, long long n) {
  long long idx = (long long)blockIdx.x * blockDim.x + threadIdx.x;
  if (idx >= n) return;
  long long e = idx / 4; int j = (int)(idx % 4);
  long long s = esrc[e], d = edst[e];
  float acc = bbt[j];
  for (int k = 0; k < C_IN; ++k) {
    acc += lig[s * C_IN + k] * Wbt[k * 4 + j];
    acc += lig[d * C_IN + k] * Wbt[(C_IN + k) * 4 + j];
  }
  out[o_bt + idx] = acc;
}

extern "C" void kernel_launch(void* const* d_in, const int* in_sizes, int n_in,
                              void* d_out, int out_size, void* d_ws, size_t ws_size,
                              hipStream_t stream) {
  (void)n_in; (void)out_size; (void)ws_size;
  const float* lig   = (const float*)d_in[0];
  const float* prot  = (const float*)d_in[1];
  const float* lpos  = (const float*)d_in[2];
  const float* ppos  = (const float*)d_in[3];
  const int*   pli   = (const int*)d_in[4];
  const int*   ppi   = (const int*)d_in[5];
  const int*   pb    = (const int*)d_in[6];
  const long long* esrc = (const long long*)d_in[7];
  const long long* edst = (const long long*)d_in[8];
  const float* W1    = (const float*)d_in[9];
  const float* b1    = (const float*)d_in[10];
  const float* gamma = (const float*)d_in[11];
  const float* beta  = (const float*)d_in[12];
  const float* Wpi   = (const float*)d_in[13];
  const float* bpi   = (const float*)d_in[14];
  const float* Wsig  = (const float*)d_in[15];
  const float* bsig  = (const float*)d_in[16];
  const float* Wmu   = (const float*)d_in[17];
  const float* bmu   = (const float*)d_in[18];
  const float* Wat   = (const float*)d_in[19];
  const float* bat   = (const float*)d_in[20];
  const float* Wbt   = (const float*)d_in[21];
  const float* bbt   = (const float*)d_in[22];

  long long P  = in_sizes[4];
  long long Tl = in_sizes[0] / C_IN;
  long long E  = in_sizes[7];
  float* ws  = (float*)d_ws;
  float* out = (float*)d_out;

  long long o_sigma = P * 10;
  long long o_mu    = 2 * P * 10;
  long long o_dist  = 3 * P * 10;
  long long o_at    = o_dist + P;
  long long o_bt    = o_at + Tl * 17;
  long long o_pb    = o_bt + E * 4;

  int numWG = (int)((P + TILE_M - 1) / TILE_M);

  k0_pack_w1<<<128, 256, 0, stream>>>(W1, ws);
  k1_gemm_stats<<<numWG, 128, 0, stream>>>(lig, prot, pli, ppi, b1, ws, P);
  k2_finalize<<<1, 256, 0, stream>>>(gamma, beta, Wpi, bpi, Wsig, bsig, Wmu, bmu,
                                     ws, numWG, P);
  k3_gemm_heads<<<numWG, 128, 0, stream>>>(lig, prot, pli, ppi, pb, b1, ws,
                                           out, P, o_sigma, o_mu, o_pb);
  k4_dist<<<(int)((P + 255) / 256), 256, 0, stream>>>(lpos, ppos, pli, ppi, out, o_dist, P);
  long long nat = Tl * 17;
  k5_atom_types<<<(int)((nat + 255) / 256), 256, 0, stream>>>(lig, Wat, bat, out, o_at, nat);
  long long nbt = E * 4;
  k6_bond_types<<<(int)((nbt + 255) / 256), 256, 0, stream>>>(lig, esrc, edst, Wbt, bbt,
                                                              out, o_bt, nbt);
}